// DynMVGaussian_31765578121901
// MI455X (gfx1250) — compile-verified
//
#include <hip/hip_runtime.h>
#include <hip/hip_bf16.h>
#include <math.h>

typedef __attribute__((ext_vector_type(2))) float v2f;
typedef __attribute__((ext_vector_type(8))) float v8f;

// ---------------------------------------------------------------- helpers
__device__ __forceinline__ float sigmoidf_(float x) {
    return 1.f / (1.f + __expf(-x));
}

__device__ __forceinline__ void quat_to_mat(float qw, float qx, float qy, float qz,
                                            float R[9]) {
    float inv = rsqrtf(qw * qw + qx * qx + qy * qy + qz * qz);
    float r = qw * inv, i = qx * inv, j = qy * inv, k = qz * inv;
    R[0] = 1.f - 2.f * (j * j + k * k); R[1] = 2.f * (i * j - k * r); R[2] = 2.f * (i * k + j * r);
    R[3] = 2.f * (i * j + k * r); R[4] = 1.f - 2.f * (i * i + k * k); R[5] = 2.f * (j * k - i * r);
    R[6] = 2.f * (i * k - j * r); R[7] = 2.f * (j * k + i * r); R[8] = 1.f - 2.f * (i * i + j * j);
}

// ---------------------------------------------------------------- prep kernels
// aug[t*M+m] = (nx, ny, nz, |n|^2)  -- the WMMA B columns
__global__ void aug_kernel(const float* __restrict__ nxyz, float4* __restrict__ aug, int tm) {
    int i = blockIdx.x * blockDim.x + threadIdx.x;
    if (i >= tm) return;
    float x = nxyz[3 * i], y = nxyz[3 * i + 1], z = nxyz[3 * i + 2];
    aug[i] = make_float4(x, y, z, x * x + y * y + z * z);
}

__global__ void count_kernel(const int* __restrict__ tidx, const int* __restrict__ sidx,
                             int* __restrict__ cnt, int S) {
    int s = blockIdx.x * blockDim.x + threadIdx.x;
    if (s >= S) return;
    atomicAdd(&cnt[tidx[sidx[s]]], 1);
}

__global__ void scan_kernel(const int* __restrict__ cnt, int* __restrict__ binOff,
                            int* __restrict__ tileStart, int* __restrict__ cursor, int T) {
    if (blockIdx.x != 0 || threadIdx.x != 0) return;
    int acc = 0, tacc = 0;
    for (int t = 0; t < T; ++t) {
        binOff[t] = acc;
        cursor[t] = acc;
        tileStart[t] = tacc;
        acc += cnt[t];
        tacc += (cnt[t] + 15) >> 4;
    }
    binOff[T] = acc;
    tileStart[T] = tacc;
}

__global__ void scatter_kernel(const int* __restrict__ tidx, const int* __restrict__ sidx,
                               int* __restrict__ cursor, int* __restrict__ binned, int S) {
    int s = blockIdx.x * blockDim.x + threadIdx.x;
    if (s >= S) return;
    int t = tidx[sidx[s]];
    int pos = atomicAdd(&cursor[t], 1);
    binned[pos] = s;
}

// ---------------------------------------------------------------- main kernel
// One wave (32 lanes) per block; each block owns a tile of 16 same-time points.
// KNN via V_WMMA_F32_16X16X4_F32:  A row p = (-2x0,-2x1,-2x2, 1),
//                                  B col n = ( n0,  n1,  n2, |n|^2)
// => D[p][n] = -2 x.n + |n|^2  (argmin == argmin of true d2).
__global__ void __launch_bounds__(32) main_kernel(
    const float* __restrict__ ray_o, const float* __restrict__ ray_d,
    const float* __restrict__ dep_base, const float* __restrict__ dep_c,
    const float* __restrict__ quat_w, const float* __restrict__ s_logit,
    const float* __restrict__ o_logit, const float* __restrict__ sph,
    const float* __restrict__ node_xyz, const float* __restrict__ node_quat,
    const int* __restrict__ support_idx, const int* __restrict__ target_t_p,
    const int* __restrict__ cnt, const int* __restrict__ binOff,
    const int* __restrict__ tileStart, const int* __restrict__ binned,
    const float4* __restrict__ aug, float* __restrict__ out,
    int S, int M, int T) {
    int lane = threadIdx.x;
    int tile = blockIdx.x;
    if (tile >= tileStart[T]) return;          // uniform exit -> EXEC stays full below

    int t = 0;                                  // which time bin does this tile belong to
    while (t < T - 1 && tile >= tileStart[t + 1]) ++t;
    int lt = tile - tileStart[t];
    int cT = cnt[t];
    int base = binOff[t] + lt * 16;
    int nval = min(16, cT - lt * 16);

    int p  = lane & 15;
    int pc = min(p, nval - 1);                  // clamp tail lanes to a valid point
    int sI = binned[base + pc];
    int gi = support_idx[sI];

    float dep = dep_base[gi] + dep_c[gi];
    float x0 = ray_o[3 * gi + 0] + ray_d[3 * gi + 0] * dep;
    float x1 = ray_o[3 * gi + 1] + ray_d[3 * gi + 1] * dep;
    float x2 = ray_o[3 * gi + 2] + ray_d[3 * gi + 2] * dep;

    bool hi = lane >= 16;                       // A: lanes 0-15 hold K0/K1, 16-31 hold K2/K3
    v2f a;
    a.x = hi ? (-2.f * x2) : (-2.f * x0);
    a.y = hi ? 1.f : (-2.f * x1);

    float bestv[8];
    int besti[8];
#pragma unroll
    for (int v = 0; v < 8; ++v) { bestv[v] = 3.402823e38f; besti[v] = 0; }

    const float4* augT = aug + (size_t)t * M;
    int nl = lane & 15;
    int nchunks = M >> 4;
    for (int c = 0; c < nchunks; ++c) {
        int n = c * 16 + nl;
        float4 nb = augT[n];
        v2f b;
        b.x = hi ? nb.z : nb.x;
        b.y = hi ? nb.w : nb.y;
        v8f acc = {};
        acc = __builtin_amdgcn_wmma_f32_16x16x4_f32(
            false, a, false, b, (short)0, acc, false, false);
#pragma unroll
        for (int v = 0; v < 8; ++v) {
            float dv = acc[v];
            if (dv < bestv[v]) { bestv[v] = dv; besti[v] = n; }
        }
    }

    // Per-row argmin: xor-butterfly across the 16 lanes of each half (masks < 16
    // never cross halves). Tie-break on lower node index (matches jnp.argmin).
#pragma unroll
    for (int v = 0; v < 8; ++v) {
#pragma unroll
        for (int m = 8; m >= 1; m >>= 1) {
            float ov = __shfl_xor(bestv[v], m, 32);
            int   oi = __shfl_xor(besti[v], m, 32);
            if (ov < bestv[v] || (ov == bestv[v] && oi < besti[v])) {
                bestv[v] = ov; besti[v] = oi;
            }
        }
    }

    __shared__ int s_attach[16];
    if (nl == 0) {
        int h = lane >> 4;                      // half 0 -> rows 0..7, half 1 -> rows 8..15
#pragma unroll
        for (int v = 0; v < 8; ++v) s_attach[v + 8 * h] = besti[v];
    }
    __syncthreads();

    if (lane >= 16 || lane >= nval) return;     // lane p finishes point p

    int attach = s_attach[lane];
    int tt = target_t_p[0];

    const float* ps = node_xyz + ((size_t)t * M + attach) * 3;
    const float* qs = node_quat + ((size_t)t * M + attach) * 4;
    const float* pt = node_xyz + ((size_t)tt * M + attach) * 3;
    const float* qt = node_quat + ((size_t)tt * M + attach) * 4;

    float Rs[9], Rt[9], Rr[9], Rd[9], fr[9];
    quat_to_mat(qs[0], qs[1], qs[2], qs[3], Rs);
    quat_to_mat(qt[0], qt[1], qt[2], qt[3], Rt);
#pragma unroll
    for (int r = 0; r < 3; ++r)
#pragma unroll
        for (int c = 0; c < 3; ++c)             // R_rel = R_tgt * R_src^T
            Rr[r * 3 + c] = Rt[r * 3 + 0] * Rs[c * 3 + 0] +
                            Rt[r * 3 + 1] * Rs[c * 3 + 1] +
                            Rt[r * 3 + 2] * Rs[c * 3 + 2];

    float dx0 = x0 - ps[0], dx1 = x1 - ps[1], dx2 = x2 - ps[2];
    float mu0 = Rr[0] * dx0 + Rr[1] * dx1 + Rr[2] * dx2 + pt[0];
    float mu1 = Rr[3] * dx0 + Rr[4] * dx1 + Rr[5] * dx2 + pt[1];
    float mu2 = Rr[6] * dx0 + Rr[7] * dx1 + Rr[8] * dx2 + pt[2];

    quat_to_mat(quat_w[4 * gi + 0], quat_w[4 * gi + 1],
                quat_w[4 * gi + 2], quat_w[4 * gi + 3], Rd);
#pragma unroll
    for (int r = 0; r < 3; ++r)
#pragma unroll
        for (int c = 0; c < 3; ++c)             // fr = R_rel @ qdir
            fr[r * 3 + c] = Rr[r * 3 + 0] * Rd[0 + c] +
                            Rr[r * 3 + 1] * Rd[3 + c] +
                            Rr[r * 3 + 2] * Rd[6 + c];

    // outputs: mu[S,3] | fr[S,3,3] | s[S,3] | o[S,1] | sph[S,27]
    float* omu  = out;
    float* ofr  = out + 3  * (size_t)S;
    float* osc  = out + 12 * (size_t)S;
    float* oop  = out + 15 * (size_t)S;
    float* osph = out + 16 * (size_t)S;

    omu[3 * (size_t)sI + 0] = mu0;
    omu[3 * (size_t)sI + 1] = mu1;
    omu[3 * (size_t)sI + 2] = mu2;
#pragma unroll
    for (int k = 0; k < 9; ++k) ofr[9 * (size_t)sI + k] = fr[k];
#pragma unroll
    for (int k = 0; k < 3; ++k)
        osc[3 * (size_t)sI + k] = sigmoidf_(s_logit[3 * gi + k]) * 0.03f;
    oop[sI] = sigmoidf_(o_logit[gi]);
    osph[27 * (size_t)sI + 0] = sph[3 * gi + 0];
    osph[27 * (size_t)sI + 1] = sph[3 * gi + 1];
    osph[27 * (size_t)sI + 2] = sph[3 * gi + 2];
#pragma unroll
    for (int k = 3; k < 27; ++k) osph[27 * (size_t)sI + k] = 0.f;
}

// ---------------------------------------------------------------- launch
extern "C" void kernel_launch(void* const* d_in, const int* in_sizes, int n_in,
                              void* d_out, int out_size, void* d_ws, size_t ws_size,
                              hipStream_t stream) {
    const float* ray_o    = (const float*)d_in[0];
    const float* ray_d    = (const float*)d_in[1];
    const float* dep_base = (const float*)d_in[2];
    const float* dep_c    = (const float*)d_in[3];
    const float* quat_w   = (const float*)d_in[4];
    const float* s_logit  = (const float*)d_in[5];
    const float* o_logit  = (const float*)d_in[6];
    const float* sph      = (const float*)d_in[7];
    const float* node_xyz = (const float*)d_in[8];
    const float* node_quat= (const float*)d_in[9];
    const int*   time_idx = (const int*)d_in[10];
    const int*   supp_idx = (const int*)d_in[11];
    const int*   target_t = (const int*)d_in[12];

    int S  = in_sizes[11];
    int TM = in_sizes[8] / 3;
    const int T = 32;
    int M = TM / T;                 // 512 for the reference shapes

    // workspace layout
    char* w = (char*)d_ws;
    int* cnt       = (int*)w;       // 32
    int* binOff    = cnt + 32;      // 33
    int* tileStart = binOff + 33;   // 33
    int* cursor    = tileStart + 33;// 32
    int* binned    = cursor + 32;   // S
    size_t augOff  = (((size_t)(130 + S)) * 4 + 15) & ~(size_t)15;
    float4* aug    = (float4*)(w + augOff);
    float* out     = (float*)d_out;

    hipMemsetAsync(cnt, 0, 32 * sizeof(int), stream);
    aug_kernel<<<(TM + 255) / 256, 256, 0, stream>>>(node_xyz, aug, TM);
    count_kernel<<<(S + 255) / 256, 256, 0, stream>>>(time_idx, supp_idx, cnt, S);
    scan_kernel<<<1, 1, 0, stream>>>(cnt, binOff, tileStart, cursor, T);
    scatter_kernel<<<(S + 255) / 256, 256, 0, stream>>>(time_idx, supp_idx, cursor, binned, S);

    int maxTiles = (S + 15) / 16 + T;   // worst-case tile count; extras exit early
    main_kernel<<<maxTiles, 32, 0, stream>>>(
        ray_o, ray_d, dep_base, dep_c, quat_w, s_logit, o_logit, sph,
        node_xyz, node_quat, supp_idx, target_t,
        cnt, binOff, tileStart, binned, aug, out, S, M, T);
}